// YOLOX_45509473468580
// MI455X (gfx1250) — compile-verified
//
#include <hip/hip_runtime.h>
#include <hip/hip_bf16.h>
#include <stdint.h>

#define NUM_CLASSES 80
#define C_DIM 85
#define N_TOTAL 33600            // 160*160 + 80*80 + 40*40
#define TOPK 500
#define CONF_THRE 0.01f
#define NMS_THRE 0.65f

typedef unsigned int u32x4 __attribute__((ext_vector_type(4)));
typedef unsigned int u32x8 __attribute__((ext_vector_type(8)));

__device__ __forceinline__ float sigmoidf_(float x) {
    return 1.0f / (1.0f + __expf(-x));
}

// One async 16-byte global->LDS copy (ASYNCcnt path).
__device__ __forceinline__ void async_copy_b128(unsigned lds_byte_addr, const void* gaddr) {
    asm volatile("global_load_async_to_lds_b128 %0, %1, off"
                 :: "v"(lds_byte_addr), "v"((unsigned long long)(uintptr_t)gaddr)
                 : "memory");
}
__device__ __forceinline__ void wait_async0() {
    asm volatile("s_wait_asynccnt 0" ::: "memory");
}

// ---------------------------------------------------------------------------
// Kernel 1: decode one FPN level.
// The whole 85 x 256 input tile (row stride HW) is DMA'd into LDS with a
// single Tensor Data Mover instruction (tensor_load_to_lds, TENSORcnt path).
// Setting tensor_dim0 = HW - t0 makes the TDM zero-fill reads past the level
// boundary for the partial last tile. Then each thread decodes one anchor.
// ---------------------------------------------------------------------------
__global__ __launch_bounds__(256)
void yolox_decode_kernel(const float* __restrict__ x, int HW, int W, float stride,
                         int anchor_base,
                         float* __restrict__ boxesA, float* __restrict__ objA,
                         float* __restrict__ ccA,    float* __restrict__ labA,
                         float* __restrict__ scoreA)
{
    __shared__ float tile[C_DIM * 256];   // 87,040 bytes

    const int tid = threadIdx.x;
    const int b   = blockIdx.y;
    const int t0  = blockIdx.x * 256;     // first hw index of this tile

    const float* src = x + (size_t)b * C_DIM * HW;
    const unsigned lbase = (unsigned)(uintptr_t)&tile[0];

    // ---- build the Tensor DMA Descriptor (block-uniform scalars) ----
    const unsigned long long gaddr = (unsigned long long)(uintptr_t)(src + t0);

    u32x4 g0;
    g0.x = 1u;                                         // count=1, user D#
    g0.y = lbase;                                      // lds_addr (bytes)
    g0.z = (unsigned)(gaddr & 0xFFFFFFFFu);            // global_addr[31:0]
    g0.w = (unsigned)((gaddr >> 32) & 0x01FFFFFFu)     // global_addr[56:32]
         | 0x80000000u;                                // type = 2 ("image")

    const unsigned td0   = (unsigned)(HW - t0);        // remaining elems in dim0
    const unsigned td1   = C_DIM;                      // 85 rows
    const unsigned tile0 = 256u;
    const unsigned tile1 = C_DIM;
    const unsigned long long str0 = (unsigned long long)(unsigned)HW;

    u32x8 g1;
    g1.s0 = 0x00020000u;                               // data_size=2 -> 4 bytes
    g1.s1 = (td0 & 0xFFFFu) << 16;                     // tensor_dim0 lo16
    g1.s2 = (td0 >> 16) | ((td1 & 0xFFFFu) << 16);     // td0 hi16 | td1 lo16
    g1.s3 = (td1 >> 16) | (tile0 << 16);               // td1 hi16 | tile_dim0
    g1.s4 = tile1 & 0xFFFFu;                           // tile_dim1 | tile_dim2=0
    g1.s5 = (unsigned)(str0 & 0xFFFFFFFFu);            // tensor_dim0_stride lo32
    g1.s6 = (unsigned)((str0 >> 32) & 0xFFFFu);        // str0 hi16 | str1 lo16=0
    g1.s7 = 0u;                                        // tensor_dim1_stride hi

    // Issue the TDM load from wave 0 only (uniform scalar branch; 2D tensor
    // form uses descriptor groups 0 and 1 only).
    const int wave_id = __builtin_amdgcn_readfirstlane((int)threadIdx.x) >> 5;
    if (wave_id == 0) {
        asm volatile("tensor_load_to_lds %0, %1" :: "s"(g0), "s"(g1) : "memory");
        asm volatile("s_wait_tensorcnt 0x0" ::: "memory");
    }
    __syncthreads();

    const int hw = t0 + tid;
    if (hw < HW) {
        const float tx = tile[0 * 256 + tid];
        const float ty = tile[1 * 256 + tid];
        const float tw = tile[2 * 256 + tid];
        const float th = tile[3 * 256 + tid];
        const float to = tile[4 * 256 + tid];

        const int gx = hw % W;
        const int gy = hw / W;
        const float cx = (tx + (float)gx) * stride;
        const float cy = (ty + (float)gy) * stride;
        const float w  = __expf(tw) * stride;
        const float h  = __expf(th) * stride;

        const float x1 = cx - 0.5f * w;
        const float y1 = cy - 0.5f * h;
        const float x2 = cx + 0.5f * w;
        const float y2 = cy + 0.5f * h;

        // class max/argmax on logits (sigmoid strictly monotone; first-max
        // tie-break matches jnp.argmax)
        float m = tile[5 * 256 + tid];
        int   mi = 0;
        #pragma unroll 4
        for (int c = 1; c < NUM_CLASSES; ++c) {
            float v = tile[(5 + c) * 256 + tid];
            if (v > m) { m = v; mi = c; }
        }

        const float objv = sigmoidf_(to);
        const float ccv  = sigmoidf_(m);
        const float tot  = objv * ccv;
        const float sc   = (tot >= CONF_THRE) ? tot : -1.0f;

        const size_t o = (size_t)b * N_TOTAL + (anchor_base + hw);
        boxesA[o * 4 + 0] = x1;
        boxesA[o * 4 + 1] = y1;
        boxesA[o * 4 + 2] = x2;
        boxesA[o * 4 + 3] = y2;
        objA[o]   = objv;
        ccA[o]    = ccv;
        labA[o]   = (float)mi;
        scoreA[o] = sc;
    }
}

// ---------------------------------------------------------------------------
// Kernel 2: per-image top-500 (lax.top_k semantics) + NMS + masked output.
// One block per image; the full 33600-float score array lives in LDS (134 KB,
// CDNA5 320KB-WGP feature), staged with async b128 copies (ASYNCcnt path).
// After selection the same LDS region is re-carved for the NMS working set.
// ---------------------------------------------------------------------------
__global__ __launch_bounds__(1024)
void yolox_topk_nms_kernel(const float* __restrict__ boxesA, const float* __restrict__ objA,
                           const float* __restrict__ ccA,    const float* __restrict__ labA,
                           const float* __restrict__ scoreA, float* __restrict__ out)
{
    __shared__ float s_sc[N_TOTAL];      // 134,400 B
    __shared__ float s_red[1024];
    __shared__ int   s_redi[1024];
    __shared__ int   s_sel[TOPK];
    __shared__ float s_val[TOPK];
    __shared__ int   s_flag;
    __shared__ float s_maxc;

    const int tid = threadIdx.x;
    const int b   = blockIdx.x;

    // ---- stage scores into LDS (33600 floats = 8400 b128 transfers) ----
    const float* ssrc = scoreA + (size_t)b * N_TOTAL;
    const unsigned sbase = (unsigned)(uintptr_t)&s_sc[0];
    for (int f = tid; f < N_TOTAL / 4; f += 1024) {
        async_copy_b128(sbase + (unsigned)(f * 16), ssrc + (size_t)f * 4);
    }
    wait_async0();
    __syncthreads();

    // ---- iterative top-500 selection (value desc, tie -> lower index) ----
    const float NEG_INF = -__builtin_inff();
    for (int k = 0; k < TOPK; ++k) {
        float bv = NEG_INF;
        int   bi = 0x7fffffff;
        for (int i = tid; i < N_TOTAL; i += 1024) {
            float v = s_sc[i];
            if (v > bv || (v == bv && i < bi)) { bv = v; bi = i; }
        }
        s_red[tid]  = bv;
        s_redi[tid] = bi;
        __syncthreads();
        for (int off = 512; off > 0; off >>= 1) {
            if (tid < off) {
                float v  = s_red[tid + off];
                int   ii = s_redi[tid + off];
                if (v > s_red[tid] || (v == s_red[tid] && ii < s_redi[tid])) {
                    s_red[tid] = v; s_redi[tid] = ii;
                }
            }
            __syncthreads();
        }
        if (tid == 0) {
            s_sel[k] = s_redi[0];
            s_val[k] = s_red[0];
            s_sc[s_redi[0]] = NEG_INF;     // remove from further rounds
        }
        __syncthreads();
    }

    // ---- re-carve s_sc for the NMS working set ----
    float* s_box  = s_sc + 0;            // 500*4
    float* s_obj  = s_sc + 2000;         // 500
    float* s_cc   = s_sc + 2500;         // 500
    float* s_lab  = s_sc + 3000;         // 500
    float* s_boff = s_sc + 3500;         // 500*4
    int*   s_sup  = (int*)(s_sc + 5500); // 500
    int*   s_keep = (int*)(s_sc + 6000); // 500

    if (tid < TOPK) {
        const int idx = s_sel[tid];
        const size_t o = (size_t)b * N_TOTAL + idx;
        s_box[tid * 4 + 0] = boxesA[o * 4 + 0];
        s_box[tid * 4 + 1] = boxesA[o * 4 + 1];
        s_box[tid * 4 + 2] = boxesA[o * 4 + 2];
        s_box[tid * 4 + 3] = boxesA[o * 4 + 3];
        s_obj[tid] = objA[o];
        s_cc[tid]  = ccA[o];
        s_lab[tid] = labA[o];
    }
    __syncthreads();

    // ---- max_c = max(where(valid, boxes, 0)) + 1 ----
    float lm = NEG_INF;
    if (tid < TOPK) {
        const bool valid = (s_val[tid] >= CONF_THRE);
        if (valid) {
            lm = fmaxf(fmaxf(s_box[tid * 4 + 0], s_box[tid * 4 + 1]),
                       fmaxf(s_box[tid * 4 + 2], s_box[tid * 4 + 3]));
        } else {
            lm = 0.0f;
        }
    }
    s_red[tid] = lm;
    __syncthreads();
    for (int off = 512; off > 0; off >>= 1) {
        if (tid < off) s_red[tid] = fmaxf(s_red[tid], s_red[tid + off]);
        __syncthreads();
    }
    if (tid == 0) s_maxc = s_red[0] + 1.0f;
    __syncthreads();
    const float max_c = s_maxc;

    if (tid < TOPK) {
        const float offv = s_lab[tid] * max_c;
        s_boff[tid * 4 + 0] = s_box[tid * 4 + 0] + offv;
        s_boff[tid * 4 + 1] = s_box[tid * 4 + 1] + offv;
        s_boff[tid * 4 + 2] = s_box[tid * 4 + 2] + offv;
        s_boff[tid * 4 + 3] = s_box[tid * 4 + 3] + offv;
        s_sup[tid]  = (s_val[tid] >= CONF_THRE) ? 0 : 1;   // suppressed = !valid
        s_keep[tid] = 0;
    }
    __syncthreads();

    // ---- sequential NMS, parallel IoU per step ----
    for (int i = 0; i < TOPK; ++i) {
        if (tid == 0) {
            int ok = s_sup[i] ? 0 : 1;
            s_keep[i] = ok;
            s_flag = ok;
        }
        __syncthreads();
        const int ok = s_flag;
        if (ok && tid < TOPK && tid > i) {
            const float bi0 = s_boff[i * 4 + 0], bi1 = s_boff[i * 4 + 1];
            const float bi2 = s_boff[i * 4 + 2], bi3 = s_boff[i * 4 + 3];
            const float bj0 = s_boff[tid * 4 + 0], bj1 = s_boff[tid * 4 + 1];
            const float bj2 = s_boff[tid * 4 + 2], bj3 = s_boff[tid * 4 + 3];
            const float ix1 = fmaxf(bi0, bj0);
            const float iy1 = fmaxf(bi1, bj1);
            const float ix2 = fminf(bi2, bj2);
            const float iy2 = fminf(bi3, bj3);
            const float inter = fmaxf(ix2 - ix1, 0.0f) * fmaxf(iy2 - iy1, 0.0f);
            const float a1 = fmaxf(bi2 - bi0, 0.0f) * fmaxf(bi3 - bi1, 0.0f);
            const float a2 = fmaxf(bj2 - bj0, 0.0f) * fmaxf(bj3 - bj1, 0.0f);
            const float iou = inter / (a1 + a2 - inter + 1e-9f);
            if (iou > NMS_THRE) s_sup[tid] = 1;
        }
        __syncthreads();
    }

    // ---- masked 7-wide output ----
    if (tid < TOPK) {
        const float kf = s_keep[tid] ? 1.0f : 0.0f;
        float* o = out + ((size_t)b * TOPK + tid) * 7;
        o[0] = s_box[tid * 4 + 0] * kf;
        o[1] = s_box[tid * 4 + 1] * kf;
        o[2] = s_box[tid * 4 + 2] * kf;
        o[3] = s_box[tid * 4 + 3] * kf;
        o[4] = s_obj[tid] * kf;
        o[5] = s_cc[tid] * kf;
        o[6] = s_lab[tid] * kf;
    }
}

extern "C" void kernel_launch(void* const* d_in, const int* in_sizes, int n_in,
                              void* d_out, int out_size, void* d_ws, size_t ws_size,
                              hipStream_t stream) {
    (void)in_sizes; (void)n_in; (void)out_size; (void)ws_size;
    const float* out8  = (const float*)d_in[0];   // (8, 85, 160, 160)
    const float* out16 = (const float*)d_in[1];   // (8, 85,  80,  80)
    const float* out32 = (const float*)d_in[2];   // (8, 85,  40,  40)

    const int B = 8;
    float* ws      = (float*)d_ws;
    float* scoreA  = ws;                               // B*N
    float* boxesA  = scoreA + (size_t)B * N_TOTAL;     // B*N*4
    float* objA    = boxesA + (size_t)B * N_TOTAL * 4; // B*N
    float* ccA     = objA   + (size_t)B * N_TOTAL;     // B*N
    float* labA    = ccA    + (size_t)B * N_TOTAL;     // B*N

    // level 8: HW=25600, 100 tiles; level 16: HW=6400, 25 tiles; level 32: HW=1600, 7 tiles
    yolox_decode_kernel<<<dim3(100, B), 256, 0, stream>>>(out8, 25600, 160, 8.0f, 0,
                                                          boxesA, objA, ccA, labA, scoreA);
    yolox_decode_kernel<<<dim3(25, B), 256, 0, stream>>>(out16, 6400, 80, 16.0f, 25600,
                                                         boxesA, objA, ccA, labA, scoreA);
    yolox_decode_kernel<<<dim3(7, B), 256, 0, stream>>>(out32, 1600, 40, 32.0f, 32000,
                                                        boxesA, objA, ccA, labA, scoreA);

    yolox_topk_nms_kernel<<<dim3(B), 1024, 0, stream>>>(boxesA, objA, ccA, labA, scoreA,
                                                        (float*)d_out);
}